// TopKRouter_80247168958768
// MI455X (gfx1250) — compile-verified
//
#include <hip/hip_runtime.h>
#include <hip/hip_bf16.h>

// MoE top-k router for MI455X (gfx1250, wave32).
//   x: [T=32768, H=2048] fp32   W: [E=8, H=2048] fp32
//   logits = x @ W^T  -> top-2 -> softmax(2) -> scores, idx, histogram
// Memory-bound: 256MB of x @ 23.3 TB/s ~= 11us floor. Gating GEMM done with
// V_WMMA_F32_16X16X4_F32 (full fp32 precision; exercises CDNA5 matrix pipe).
//
// B-matrix trick: lanes 8..15 replicate experts 0..7 (W[n&7]) instead of
// being zeroed, so D columns 8..15 are exact duplicates of 0..7. The epilogue
// only reads columns 0..7, so no per-iteration masking VALU is needed.

#define ROUTER_T 32768
#define ROUTER_H 2048
#define ROUTER_E 8
#define TOPK 2

typedef __attribute__((ext_vector_type(2))) float v2f;
typedef __attribute__((ext_vector_type(8))) float v8f;

__global__ void zero_hist_kernel(float* hist) {
    if (threadIdx.x < ROUTER_E) hist[threadIdx.x] = 0.0f;
}

// 256 threads = 8 wave32 per block. Each wave computes a 16-token x 16-expert
// logits tile (expert columns 8..15 duplicate 0..7, ignored in the epilogue).
__global__ __launch_bounds__(256) void moe_router_kernel(
    const float* __restrict__ x, const float* __restrict__ W,
    float* __restrict__ out)
{
    float* scores = out;                        // [T,2] fp32
    float* idxOut = out + ROUTER_T * TOPK;      // [T,2] indices stored as float
    float* hist   = out + ROUTER_T * TOPK * 2;  // [8]

    // Per-wave 16x16 logits tile, row stride 17 to avoid LDS bank conflicts.
    __shared__ float tile[8][16 * 17];
    __shared__ float lhist[ROUTER_E];

    const int lane = threadIdx.x & 31;
    const int wave = threadIdx.x >> 5;
    const int waveTokenBase = (blockIdx.x * 8 + wave) * 16;

    // ---- WMMA fragment addressing (V_WMMA_F32_16X16X4_F32) ----
    // A (16x4, M x K): lanes 0-15 hold M=lane, VGPR0/1 = K0/K1;
    //                  lanes 16-31 hold M=lane-16, VGPR0/1 = K2/K3.
    // B (4x16, K x N): symmetric layout with N in place of M.
    const int mrow  = lane & 15;       // token row (A) / expert col (B)
    const int khalf = (lane >> 4) * 2; // K offset within the 4-wide step

    const float* aP = x + (size_t)(waveTokenBase + mrow) * ROUTER_H + khalf;
    const float* bP = W + (size_t)(mrow & (ROUTER_E - 1)) * ROUTER_H + khalf;

    v8f c = {};
    for (int k = 0; k < ROUTER_H; k += 16) {
        // Prefetch the streaming operand ~1KB ahead (global_prefetch_b8).
        __builtin_prefetch(aP + k + 256, 0, 1);
#pragma unroll
        for (int u = 0; u < 4; ++u) {
            v2f a = *(const v2f*)(aP + k + 4 * u);   // global_load_b64
            v2f b = *(const v2f*)(bP + k + 4 * u);   // global_load_b64 (L2-hot)
            // 8 args: (neg_a, A, neg_b, B, c_mod, C, reuse_a, reuse_b)
            c = __builtin_amdgcn_wmma_f32_16x16x4_f32(
                false, a, false, b, (short)0, c, false, false);
        }
    }

    // ---- Spill D tile to LDS: D layout is lanes 0-15 -> N=lane, M=r;
    //      lanes 16-31 -> N=lane-16, M=r+8.  tile[M][N] with stride 17. ----
    {
        float* myTile = &tile[wave][0];
        const int ncol = lane & 15;
        const int moff = (lane >> 4) * 8;
#pragma unroll
        for (int r = 0; r < 8; ++r)
            myTile[(r + moff) * 17 + ncol] = c[r];
    }

    if (threadIdx.x < ROUTER_E) lhist[threadIdx.x] = 0.0f;
    __syncthreads();

    // ---- Per-token epilogue: top-2 (ties -> lowest index, like lax.top_k),
    //      2-way softmax, LDS histogram. One thread per token (128/block). ----
    if (threadIdx.x < 128) {
        const int localTok = threadIdx.x;
        const int srcWave  = localTok >> 4;
        const int r        = localTok & 15;
        const float* row   = &tile[srcWave][r * 17];

        float best = row[0]; int bi = 0;
#pragma unroll
        for (int e = 1; e < ROUTER_E; ++e) {
            float v = row[e];
            if (v > best) { best = v; bi = e; }
        }
        float second = -__builtin_inff(); int si = 0;
#pragma unroll
        for (int e = 0; e < ROUTER_E; ++e) {
            if (e == bi) continue;
            float v = row[e];
            if (v > second) { second = v; si = e; }
        }

        const float p0 = 1.0f / (1.0f + expf(second - best));
        const float p1 = 1.0f - p0;

        const int t = blockIdx.x * 128 + localTok;
        scores[2 * t + 0] = p0;
        scores[2 * t + 1] = p1;
        idxOut[2 * t + 0] = (float)bi;
        idxOut[2 * t + 1] = (float)si;

        atomicAdd(&lhist[bi], 1.0f);   // ds_add_f32
        atomicAdd(&lhist[si], 1.0f);
    }
    __syncthreads();

    if (threadIdx.x < ROUTER_E)
        atomicAdd(&hist[threadIdx.x], lhist[threadIdx.x]);  // global_atomic_add_f32
}

extern "C" void kernel_launch(void* const* d_in, const int* in_sizes, int n_in,
                              void* d_out, int out_size, void* d_ws, size_t ws_size,
                              hipStream_t stream) {
    const float* x = (const float*)d_in[0];   // [T, H]
    const float* W = (const float*)d_in[1];   // [E, H]
    float* out = (float*)d_out;

    const int T = in_sizes[0] / ROUTER_H;     // 32768

    // Zero the histogram region first (stream-ordered before the atomics).
    zero_hist_kernel<<<1, 32, 0, stream>>>(out + (size_t)T * TOPK * 2);

    // 128 tokens per block (8 waves x 16 tokens).
    const int blocks = T / 128;
    moe_router_kernel<<<blocks, 256, 0, stream>>>(x, W, out);
}